// PillarVFE_41034117546487
// MI455X (gfx1250) — compile-verified
//
#include <hip/hip_runtime.h>
#include <math.h>

typedef float v2f __attribute__((ext_vector_type(2)));
typedef float v8f __attribute__((ext_vector_type(8)));

#define PP 32
#define COUT 64
static constexpr float kVX = 0.16f, kVY = 0.16f, kVZ = 4.0f;
static constexpr float kXOFF = kVX * 0.5f + 0.0f;
static constexpr float kYOFF = kVY * 0.5f - 39.68f;
static constexpr float kZOFF = kVZ * 0.5f - 3.0f;
static constexpr float kEPS = 0.001f;

// workspace layout (floats)
#define MOM_OFF   0    // 65: 10 first moments + 55 upper-tri second moments
#define SCALE_OFF 80   // 64
#define SHIFT_OFF 144  // 64

__global__ void vfe_zero(float* __restrict__ ws) {
  if (threadIdx.x < 65) ws[MOM_OFF + threadIdx.x] = 0.0f;
}

// ---------------- Pass A: feature moments (memory-bound, 51 MB read) -------
__global__ void __launch_bounds__(256) vfe_moments(
    const float* __restrict__ vf, const int* __restrict__ npts,
    const int* __restrict__ coords, float* __restrict__ ws, int V) {
  const int lane = threadIdx.x & 31;
  const int wib  = threadIdx.x >> 5;  // wave in block (8 waves)
  const int wave = blockIdx.x * 8 + wib;
  const int nwav = gridDim.x * 8;

  float acc[65];
#pragma unroll
  for (int t = 0; t < 65; ++t) acc[t] = 0.0f;

  for (int v = wave; v < V; v += nwav) {
    const float4 p4 = ((const float4*)vf)[v * PP + lane];
    const int cnt = npts[v];
    // mean of xyz over ALL 32 points, divided by cnt (reference semantics)
    float sx = p4.x, sy = p4.y, sz = p4.z;
#pragma unroll
    for (int m = 16; m >= 1; m >>= 1) {
      sx += __shfl_xor(sx, m, 32);
      sy += __shfl_xor(sy, m, 32);
      sz += __shfl_xor(sz, m, 32);
    }
    const float inv = 1.0f / (float)cnt;
    const float mx = sx * inv, my = sy * inv, mz = sz * inv;
    const float cx = (float)coords[v * 4 + 3] * kVX + kXOFF;
    const float cy = (float)coords[v * 4 + 2] * kVY + kYOFF;
    const float cz = (float)coords[v * 4 + 1] * kVZ + kZOFF;
    const float msk = (lane < cnt) ? 1.0f : 0.0f;
    float f[10];
    f[0] = p4.x * msk; f[1] = p4.y * msk; f[2] = p4.z * msk; f[3] = p4.w * msk;
    f[4] = (p4.x - mx) * msk; f[5] = (p4.y - my) * msk; f[6] = (p4.z - mz) * msk;
    f[7] = (p4.x - cx) * msk; f[8] = (p4.y - cy) * msk; f[9] = (p4.z - cz) * msk;
#pragma unroll
    for (int i = 0; i < 10; ++i) acc[i] += f[i];
    int t = 10;
#pragma unroll
    for (int i = 0; i < 10; ++i)
#pragma unroll
      for (int j = i; j < 10; ++j) acc[t++] += f[i] * f[j];
  }

  // wave butterfly reduce each accumulator
#pragma unroll
  for (int t = 0; t < 65; ++t)
#pragma unroll
    for (int m = 16; m >= 1; m >>= 1) acc[t] += __shfl_xor(acc[t], m, 32);

  __shared__ float part[8][72];
  if (lane == 0) {
#pragma unroll
    for (int t = 0; t < 65; ++t) part[wib][t] = acc[t];
  }
  __syncthreads();
  if (threadIdx.x < 65) {
    float s = 0.0f;
#pragma unroll
    for (int w = 0; w < 8; ++w) s += part[w][threadIdx.x];
    atomicAdd(&ws[MOM_OFF + threadIdx.x], s);
  }
}

// ---------------- Pass B: per-channel BN scale/shift from moments ----------
__global__ void vfe_stats(const float* __restrict__ W,
                          const float* __restrict__ gamma,
                          const float* __restrict__ beta,
                          float* __restrict__ ws, int V) {
  const int o = threadIdx.x;
  if (o >= COUT) return;
  const float N = (float)V * (float)PP;
  float w[10];
#pragma unroll
  for (int i = 0; i < 10; ++i) w[i] = W[o * 10 + i];
  float mean = 0.0f;
#pragma unroll
  for (int i = 0; i < 10; ++i) mean += ws[MOM_OFF + i] * w[i];
  mean /= N;
  float ex2 = 0.0f;
  int t = 10;
#pragma unroll
  for (int i = 0; i < 10; ++i)
#pragma unroll
    for (int j = i; j < 10; ++j) {
      const float m = ws[MOM_OFF + t++];
      ex2 += ((i == j) ? 1.0f : 2.0f) * w[i] * w[j] * m;
    }
  ex2 /= N;
  const float var = ex2 - mean * mean;
  const float sc = rsqrtf(var + kEPS) * gamma[o];
  ws[SCALE_OFF + o] = sc;
  ws[SHIFT_OFF + o] = beta[o] - mean * sc;
}

// ------- Pass C: fused WMMA GEMM + BN + ReLU + max over P (wave/voxel) -----
__global__ void __launch_bounds__(256) vfe_gemm(
    const float* __restrict__ vf, const int* __restrict__ npts,
    const int* __restrict__ coords, const float* __restrict__ W,
    const float* __restrict__ ws, float* __restrict__ out, int V) {
  __shared__ float Blds[12][64];      // W^T, K padded 10->12 with zeros
  __shared__ float Alds[8][32][12];   // per-wave 32x12 feature tile

  const int tid = threadIdx.x, lane = tid & 31, wib = tid >> 5;
  const int m16 = lane & 15, kh = lane >> 4;

  for (int idx = tid; idx < 12 * 64; idx += 256) {
    const int k = idx >> 6, n = idx & 63;
    Blds[k][n] = (k < 10) ? W[n * 10 + k] : 0.0f;
  }
  __syncthreads();

  // hoist B fragments (invariant over voxels): VGPR0=K0/K2, VGPR1=K1/K3
  v2f bf[4][3];
#pragma unroll
  for (int nt = 0; nt < 4; ++nt)
#pragma unroll
    for (int kt = 0; kt < 3; ++kt) {
      bf[nt][kt].x = Blds[kt * 4 + 2 * kh + 0][nt * 16 + m16];
      bf[nt][kt].y = Blds[kt * 4 + 2 * kh + 1][nt * 16 + m16];
    }
  float sc[4], sh[4];
#pragma unroll
  for (int nt = 0; nt < 4; ++nt) {
    sc[nt] = ws[SCALE_OFF + nt * 16 + m16];
    sh[nt] = ws[SHIFT_OFF + nt * 16 + m16];
  }

  float* Aw = &Alds[wib][0][0];
  const int wave = blockIdx.x * 8 + wib;
  const int nwav = gridDim.x * 8;

  for (int v = wave; v < V; v += nwav) {
    const float4 p4 = ((const float4*)vf)[v * PP + lane];
    const int cnt = npts[v];
    float sx = p4.x, sy = p4.y, sz = p4.z;
#pragma unroll
    for (int m = 16; m >= 1; m >>= 1) {
      sx += __shfl_xor(sx, m, 32);
      sy += __shfl_xor(sy, m, 32);
      sz += __shfl_xor(sz, m, 32);
    }
    const float inv = 1.0f / (float)cnt;
    const float mx = sx * inv, my = sy * inv, mz = sz * inv;
    const float cx = (float)coords[v * 4 + 3] * kVX + kXOFF;
    const float cy = (float)coords[v * 4 + 2] * kVY + kYOFF;
    const float cz = (float)coords[v * 4 + 1] * kVZ + kZOFF;
    const float msk = (lane < cnt) ? 1.0f : 0.0f;

    float4* Arow = (float4*)(Aw + lane * 12);
    Arow[0] = make_float4(p4.x * msk, p4.y * msk, p4.z * msk, p4.w * msk);
    Arow[1] = make_float4((p4.x - mx) * msk, (p4.y - my) * msk,
                          (p4.z - mz) * msk, (p4.x - cx) * msk);
    Arow[2] = make_float4((p4.y - cy) * msk, (p4.z - cz) * msk, 0.0f, 0.0f);
    __asm__ volatile("s_wait_dscnt 0" ::: "memory");  // DS write->read fence

    float vmax[4];
#pragma unroll
    for (int nt = 0; nt < 4; ++nt) vmax[nt] = -__builtin_inff();

#pragma unroll
    for (int mt = 0; mt < 2; ++mt) {
      // A fragments: lane holds row M=m16, VGPR0=K0/K2, VGPR1=K1/K3 per half
      v2f af[3];
#pragma unroll
      for (int kt = 0; kt < 3; ++kt) {
        const float* ap = Aw + (mt * 16 + m16) * 12 + kt * 4 + 2 * kh;
        af[kt].x = ap[0];
        af[kt].y = ap[1];
      }
#pragma unroll
      for (int nt = 0; nt < 4; ++nt) {
        v8f acc = {0.f, 0.f, 0.f, 0.f, 0.f, 0.f, 0.f, 0.f};
#pragma unroll
        for (int kt = 0; kt < 3; ++kt) {
          acc = __builtin_amdgcn_wmma_f32_16x16x4_f32(
              false, af[kt], false, bf[nt][kt], (short)0, acc, false, false);
        }
        // C/D layout: VGPR r -> M = r + 8*kh (+16*mt); BN + ReLU + local max
#pragma unroll
        for (int r = 0; r < 8; ++r) {
          float x = acc[r] * sc[nt] + sh[nt];
          x = fmaxf(x, 0.0f);
          vmax[nt] = fmaxf(vmax[nt], x);
        }
      }
    }
    // lane^16 partner holds the other 16 M values for the same channel n
#pragma unroll
    for (int nt = 0; nt < 4; ++nt) {
      const float m = fmaxf(vmax[nt], __shfl_xor(vmax[nt], 16, 32));
      if (lane < 16) out[v * 64 + nt * 16 + lane] = m;
    }
  }
}

extern "C" void kernel_launch(void* const* d_in, const int* in_sizes, int n_in,
                              void* d_out, int out_size, void* d_ws, size_t ws_size,
                              hipStream_t stream) {
  const float* vf     = (const float*)d_in[0];
  const int*   np     = (const int*)d_in[1];
  const int*   coords = (const int*)d_in[2];
  const float* W      = (const float*)d_in[3];
  const float* gamma  = (const float*)d_in[4];
  const float* beta   = (const float*)d_in[5];
  float* out = (float*)d_out;
  float* ws  = (float*)d_ws;
  const int V = in_sizes[1];  // voxel_num_points is (V,)

  vfe_zero<<<1, 128, 0, stream>>>(ws);
  vfe_moments<<<512, 256, 0, stream>>>(vf, np, coords, ws, V);
  vfe_stats<<<1, 64, 0, stream>>>(W, gamma, beta, ws, V);
  vfe_gemm<<<2048, 256, 0, stream>>>(vf, np, coords, W, ws, out, V);
}